// KalmanFilterWrapper_11811160064088
// MI455X (gfx1250) — compile-verified
//
#include <hip/hip_runtime.h>
#include <stdint.h>

// ---------------------------------------------------------------------------
// Kalman filter, BATCH=4096 trajectories x SEQ_LEN=4096, 2D measurements.
// Gains are batch-invariant -> precompute; x/y decouple into two 2-state
// filters. Chunked affine scan (L=16) turns per-chunk work into
// preds = G_c(16x16) @ Z(16x16 batch tile) + U_c @ x_entry  ==> WMMA f32.
// Z tiles / pred tiles move via the Tensor Data Mover (2D tile descriptors,
// double buffered, s_wait_tensorcnt pipelining).
// ---------------------------------------------------------------------------

typedef float v2f __attribute__((ext_vector_type(2)));
typedef float v8f __attribute__((ext_vector_type(8)));
typedef unsigned int v4u __attribute__((ext_vector_type(4)));
typedef int v4i __attribute__((ext_vector_type(4)));
typedef int v8i __attribute__((ext_vector_type(8)));

#define BATCH   4096
#define TLEN    4096
#define LCH     16
#define NC      (TLEN / LCH)      // 256 chunks
#define TILE_B  16
#define NTILES  (BATCH / TILE_B)  // 256 workgroups
#define ROWSTR  (TLEN * 2)        // elements per batch row (8192)

// workspace layout (float offsets)
#define WS_K0 0
#define WS_K1 (WS_K0 + TLEN)
#define WS_G  (WS_K1 + TLEN)             // NC * 256
#define WS_U  (WS_G + NC * 256)          // NC * 32
#define WS_W  (WS_U + NC * 32)           // NC * 32
#define WS_M  (WS_W + NC * 32)           // NC * 4

#define ZROW 33        // padded LDS row (32 data dwords + 1 pad via TDM pad)
#define ZBUF (16 * ZROW)

#if defined(__has_builtin)
#if __has_builtin(__builtin_amdgcn_tensor_load_to_lds) && \
    __has_builtin(__builtin_amdgcn_tensor_store_from_lds) && \
    __has_builtin(__builtin_amdgcn_s_wait_tensorcnt)
#define HAVE_TDM 1
#endif
#endif
#ifndef HAVE_TDM
#define HAVE_TDM 0
#endif

// ---------------------------------------------------------------------------
// Kernel A1: Riccati gains k0[t], k1[t] for the 2-state (pos,vel) filter.
// Serial, but exact fixed-point early-exit: once P_{t+1}==P_t bit-exactly,
// all later gains are identical constants (stores only, no dep chain).
// ---------------------------------------------------------------------------
__global__ __launch_bounds__(32) void kf_gains(float* __restrict__ ws) {
    if (threadIdx.x != 0) return;
    float* K0 = ws + WS_K0;
    float* K1 = ws + WS_K1;
    const float q = 1e-5f, r = 0.1f;
    float p00 = 1.f, p01 = 0.f, p11 = 1.f;
    float k0 = 0.f, k1 = 0.f;
    bool conv = false;
    for (int t = 0; t < TLEN; ++t) {
        if (!conv) {
            // predict: P' = F P F^T + qI   (F = [[1,1],[0,1]])
            float a00 = p00 + 2.f * p01 + p11 + q;
            float a01 = p01 + p11;
            float a11 = p11 + q;
            // update: S = a00 + r; K = P'H^T/S; P = (I-KH)P'
            float S   = a00 + r;
            float nk0 = a00 / S;
            float nk1 = a01 / S;
            float n00 = (1.f - nk0) * a00;
            float n01 = (1.f - nk0) * a01;
            float n11 = a11 - nk1 * a01;
            conv = (n00 == p00) & (n01 == p01) & (n11 == p11) &
                   (nk0 == k0) & (nk1 == k1);
            p00 = n00; p01 = n01; p11 = n11; k0 = nk0; k1 = nk1;
        }
        K0[t] = k0;
        K1[t] = k1;
    }
}

// ---------------------------------------------------------------------------
// Kernel A2: per-chunk coefficient matrices (batch-invariant).
//  G[t][s] = effect of z_s on pred_t (s<=t), U[t] = effect of entry state,
//  M = chunk transition (2x2), W[s] = effect of z_s on exit state.
// One thread per chunk (256 threads).
// ---------------------------------------------------------------------------
__global__ __launch_bounds__(32) void kf_chunk_coeffs(float* __restrict__ ws) {
    int c = blockIdx.x * 32 + threadIdx.x;
    if (c >= NC) return;
    const float* K0 = ws + WS_K0;
    const float* K1 = ws + WS_K1;
    float* G = ws + WS_G + c * (LCH * LCH);
    float* U = ws + WS_U + c * (LCH * 2);
    float* W = ws + WS_W + c * (LCH * 2);
    float* M = ws + WS_M + c * 4;

    float ex[LCH], ev[LCH];
    float m00 = 1.f, m01 = 0.f, m10 = 0.f, m11 = 1.f;
    for (int t = 0; t < LCH; ++t) {
        float k0 = K0[c * LCH + t], k1 = K1[c * LCH + t];
        // x_{t+1} = A x_t + b z ; A=[[1-k0,1-k0],[-k1,1-k1]], b=[k0,k1]
        float a00 = 1.f - k0, a01 = 1.f - k0;
        float a10 = -k1,      a11 = 1.f - k1;
        for (int s = 0; s < t; ++s) {
            float nx = a00 * ex[s] + a01 * ev[s];
            float nv = a10 * ex[s] + a11 * ev[s];
            ex[s] = nx; ev[s] = nv;
        }
        ex[t] = k0; ev[t] = k1;
        float t00 = a00 * m00 + a01 * m10;
        float t01 = a00 * m01 + a01 * m11;
        float t10 = a10 * m00 + a11 * m10;
        float t11 = a10 * m01 + a11 * m11;
        m00 = t00; m01 = t01; m10 = t10; m11 = t11;
        U[t * 2 + 0] = m00;
        U[t * 2 + 1] = m01;
        for (int s = 0; s < LCH; ++s) G[t * LCH + s] = (s <= t) ? ex[s] : 0.f;
    }
    M[0] = m00; M[1] = m01; M[2] = m10; M[3] = m11;
    for (int s = 0; s < LCH; ++s) { W[s * 2 + 0] = ex[s]; W[s * 2 + 1] = ev[s]; }
}

// ---------------------------------------------------------------------------
// TDM descriptor builder: 2D tile, 16 rows x 32 f32, row stride ROWSTR elems.
// Load path adds 1-dword LDS pad per 32-dword row (bank-conflict-free reads).
// amdgpu-toolchain (clang-23) 6-arg builtin form:
//   (v4u g0, v8i g1, v4i g2, v4i g3, v8i g4, i32 cpol)
// ---------------------------------------------------------------------------
#if HAVE_TDM
__device__ __forceinline__ void tdm_issue(unsigned lds_off,
                                          unsigned long long ga,
                                          unsigned cfg, bool is_store) {
    v4u g0;
    g0[0] = 1u;                                        // count=1, user desc
    g0[1] = lds_off;                                   // LDS byte address
    g0[2] = (unsigned)(ga & 0xFFFFFFFFull);            // global addr lo
    g0[3] = (unsigned)((ga >> 32) & 0x01FFFFFFull) | (2u << 30); // hi | type=2
    v8i g1;
    g1[0] = (int)cfg;                                  // data_size=4B, pad cfg
    g1[1] = (int)((ROWSTR & 0xFFFFu) << 16);           // tensor_dim0 lo16
    g1[2] = (int)((ROWSTR >> 16) | (16u << 16));       // dim0 hi | tensor_dim1
    g1[3] = (int)(32u << 16);                          // tile_dim0 = 32 elems
    g1[4] = (int)16u;                                  // tile_dim1 = 16 rows
    g1[5] = (int)ROWSTR;                               // tensor_dim0_stride lo
    g1[6] = 0;
    g1[7] = 0;
    v4i gz = {0, 0, 0, 0};
    v8i gz8 = {0, 0, 0, 0, 0, 0, 0, 0};
    if (is_store)
        __builtin_amdgcn_tensor_store_from_lds(g0, g1, gz, gz, gz8, 0);
    else
        __builtin_amdgcn_tensor_load_to_lds(g0, g1, gz, gz, gz8, 0);
}
#define TDM_CFG_LOAD  ((2u << 16) | (1u << 20) | (4u << 22)) /* 4B, pad 1dw/32dw */
#define TDM_CFG_STORE (2u << 16)                             /* 4B, no pad */
#endif

__device__ __forceinline__ unsigned lds_lo32(const void* p) {
    // LDS aperture: low 32 bits of a generic shared pointer == LDS byte offset
    return (unsigned)(unsigned long long)(uintptr_t)p;
}

// ---------------------------------------------------------------------------
// Per-chunk compute: acc = G_c @ Z_tile via 4x v_wmma_f32_16x16x4_f32,
// then pred = acc + U_c @ x_entry; carry update x' = M_c x + W_c @ z.
// ---------------------------------------------------------------------------
__device__ __forceinline__ void compute_chunk(
    int c, int d, int n, int hi,
    const float* __restrict__ Gw, const float* __restrict__ Uw,
    const float* __restrict__ Ww, const float* __restrict__ Mw,
    const float* __restrict__ zl,   // LDS z tile, ZROW-stride rows
    float* __restrict__ pl,         // LDS pred tile, 32-stride rows (linear)
    float (*carry)[16][2]) {
    const float* Gc = Gw + c * 256;
    v8f acc = {0.f, 0.f, 0.f, 0.f, 0.f, 0.f, 0.f, 0.f};
#pragma unroll
    for (int kk = 0; kk < 4; ++kk) {
        int ka = kk * 4 + hi * 2;   // A: vgpr0 holds K=ka, vgpr1 K=ka+1
        v2f a, b;
        a.x = Gc[n * 16 + ka];
        a.y = Gc[n * 16 + ka + 1];
        b.x = zl[n * ZROW + ka * 2 + d];        // B[k][n] = z[t=k, batch n]
        b.y = zl[n * ZROW + (ka + 1) * 2 + d];
        acc = __builtin_amdgcn_wmma_f32_16x16x4_f32(
            false, a, false, b, (short)0, acc, false, false);
    }
    float xp = carry[d][n][0];
    float xv = carry[d][n][1];
    const float* Uc = Uw + c * 32;
#pragma unroll
    for (int rr = 0; rr < 8; ++rr) {
        int m = rr + hi * 8;        // D vgpr rr: lanes0-15 M=rr, lanes16-31 M=rr+8
        float pred = acc[rr] + Uc[m * 2] * xp + Uc[m * 2 + 1] * xv;
        pl[n * 32 + m * 2 + d] = pred;
    }
    const float* Wc = Ww + c * 32;
    float s = 0.f;
#pragma unroll
    for (int k = 0; k < LCH; ++k)
        s += Wc[k * 2 + hi] * zl[n * ZROW + k * 2 + d];
    const float* Mc = Mw + c * 4;
    float nc = Mc[hi * 2] * xp + Mc[hi * 2 + 1] * xv + s;
    carry[d][n][hi] = nc;           // same-wave DS ordering: reads precede write
}

// ---------------------------------------------------------------------------
// Main kernel: one workgroup (2 waves) per 16-batch tile, sequential over
// 256 chunks, TDM double-buffered Z loads + TDM pred stores.
// ---------------------------------------------------------------------------
__global__ __launch_bounds__(64) void kf_main(const float* __restrict__ Z,
                                              float* __restrict__ out,
                                              const float* __restrict__ ws) {
    __shared__ float zb[2][ZBUF];      // [buf][row n * 33 + (t*2+d)]
    __shared__ float pb[2][16 * 32];   // linear tile for TDM store
    __shared__ float carry[2][16][2];  // [dim][batch][pos,vel]

    const int lane = threadIdx.x & 31;
    const int wave = threadIdx.x >> 5;
    const int d = wave;                // wave 0 -> x dim, wave 1 -> y dim
    const int n = lane & 15;
    const int hi = lane >> 4;
    const int b0 = blockIdx.x * TILE_B;

    const float* Gw = ws + WS_G;
    const float* Uw = ws + WS_U;
    const float* Ww = ws + WS_W;
    const float* Mw = ws + WS_M;

    // carry init: x0 = [z0, 0] per dimension
    carry[d][n][hi] = hi ? 0.f : Z[(size_t)(b0 + n) * ROWSTR + d];

#if HAVE_TDM
    if (wave == 0) {
        tdm_issue(lds_lo32(&zb[0][0]),
                  (unsigned long long)(uintptr_t)(Z + (size_t)b0 * ROWSTR),
                  TDM_CFG_LOAD, false);
        __builtin_amdgcn_s_wait_tensorcnt(0);
    }
    __syncthreads();

    for (int c = 0; c < NC; ++c) {
        const int cb = c & 1;
        if (wave == 0) {
            if (c + 1 < NC) {
                tdm_issue(lds_lo32(&zb[(c + 1) & 1][0]),
                          (unsigned long long)(uintptr_t)
                              (Z + (size_t)b0 * ROWSTR + (size_t)(c + 1) * 32),
                          TDM_CFG_LOAD, false);
                // in-order TENSORcnt: <=2 pending => L(c) and all older done
                __builtin_amdgcn_s_wait_tensorcnt(2);
            } else {
                __builtin_amdgcn_s_wait_tensorcnt(1);
            }
        }
        __syncthreads();

        compute_chunk(c, d, n, hi, Gw, Uw, Ww, Mw, &zb[cb][0], &pb[cb][0], carry);

        __syncthreads();
        if (wave == 0) {
            tdm_issue(lds_lo32(&pb[cb][0]),
                      (unsigned long long)(uintptr_t)
                          (out + (size_t)b0 * ROWSTR + (size_t)c * 32),
                      TDM_CFG_STORE, true);
        }
        // final store drains via implicit wait at s_endpgm
    }
#else
    // fallback: synchronous cooperative copies (still WMMA compute)
    const int fn = threadIdx.x >> 2;   // 0..15 batch row
    const int fq = threadIdx.x & 3;    // quarter of row
    for (int c = 0; c < NC; ++c) {
        __syncthreads();
        {
            const float* src = Z + (size_t)(b0 + fn) * ROWSTR + (size_t)c * 32 + fq * 8;
            float* dst = &zb[0][fn * ZROW + fq * 8];
#pragma unroll
            for (int j = 0; j < 8; ++j) dst[j] = src[j];
        }
        __syncthreads();
        compute_chunk(c, d, n, hi, Gw, Uw, Ww, Mw, &zb[0][0], &pb[0][0], carry);
        __syncthreads();
        {
            float* dst = out + (size_t)(b0 + fn) * ROWSTR + (size_t)c * 32 + fq * 8;
            const float* src = &pb[0][fn * 32 + fq * 8];
#pragma unroll
            for (int j = 0; j < 8; ++j) dst[j] = src[j];
        }
    }
#endif
}

// ---------------------------------------------------------------------------
extern "C" void kernel_launch(void* const* d_in, const int* in_sizes, int n_in,
                              void* d_out, int out_size, void* d_ws, size_t ws_size,
                              hipStream_t stream) {
    const float* Z = (const float*)d_in[0];
    float* out = (float*)d_out;
    float* ws = (float*)d_ws;   // needs (2*4096 + 256*(256+32+32+4)) floats ~ 357 KB

    kf_gains<<<1, 32, 0, stream>>>(ws);
    kf_chunk_coeffs<<<NC / 32, 32, 0, stream>>>(ws);
    kf_main<<<NTILES, 64, 0, stream>>>(Z, out, ws);
}